// Conv2dTriton_9002251453192
// MI455X (gfx1250) — compile-verified
//
#include <hip/hip_runtime.h>

// Depthwise causal conv (N=2, C=16, H=W=2048, R=6, S=11, pad_h=5 causal, pad_w=5 centered)
// y[z,p,q] = [p>=q] * sum_{r,s} w[c,r,s] * ([i>=j]*x[z, p+r-5, q+s-5])
//
// Mapped onto V_WMMA_F32_16X16X4_F32 (full fp32 matrix pipe):
//   y[p, Q+j] = sum_r sum_u  A_r(m,u) * B_r(u,j)
//   A_r(m,u) = xmasked[p0+m-5+r, Q-5+u]      (16 x 28 slab of LDS window)
//   B_r(u,j) = w[r, u-j]  if 0<=u-j<=10      (banded Toeplitz, K=28 -> 7 K4-chunks)
// One wave computes a 16x64 output supertile (4 q-subtiles share the LDS window
// and the B fragments): 6*7*4 = 168 WMMAs per wave.

typedef __attribute__((ext_vector_type(2))) float v2f;
typedef __attribute__((ext_vector_type(8))) float v8f;

#define Hc 2048
#define Wc 2048
#define Cc 16
#define Nb 2
#define Rr 6
#define Ss 11

// LDS input window: rows p0-5..p0+15 (21), cols q0-5..q0+70 (76 used).
// Row stride 78 floats: even -> b64 LDS loads stay 8B aligned; 78 mod 64 = 14
// makes the 16 consecutive rows read by a half-wave hit distinct banks.
#define WROWS 21
#define WCOLS 76
#define WSTR  78

__global__ __launch_bounds__(32)
void dwconv_causal_wmma(const float* __restrict__ x,
                        const float* __restrict__ wgt,
                        float* __restrict__ y)
{
    const int lane = threadIdx.x;      // 0..31, wave32
    const int col  = lane & 15;        // N (or M) index within fragment
    const int half = lane >> 4;        // K-half selector

    const int z  = blockIdx.z;         // plane = b*C + c
    const int c  = z & (Cc - 1);
    const int p0 = blockIdx.y * 16;    // output row tile
    const int q0 = blockIdx.x * 64;    // output col supertile (4 x 16)

    const float* __restrict__ xp = x + (size_t)z * Hc * Wc;
    float*       __restrict__ yp = y + (size_t)z * Hc * Wc;

    __shared__ float xs[WROWS * WSTR];   // masked input window
    __shared__ float lw[Rr * Ss];        // this channel's 66 weights

    v8f acc[4];
#pragma unroll
    for (int qt = 0; qt < 4; ++qt)
#pragma unroll
        for (int e = 0; e < 8; ++e) acc[qt][e] = 0.0f;

    // Tiles strictly above the diagonal are fully masked: just write zeros.
    const bool skip = (p0 + 15) < q0;

    if (!skip) {
        // --- stage weights into LDS ---
        for (int i = lane; i < Rr * Ss; i += 32)
            lw[i] = wgt[c * (Rr * Ss) + i];

        // --- stage causally-masked input window into LDS ---
        for (int idx = lane; idx < WROWS * WCOLS; idx += 32) {
            int i  = idx / WCOLS;
            int u  = idx - i * WCOLS;
            int ig = p0 - 5 + i;       // global input row (<= 2047 always)
            int jg = q0 - 5 + u;       // global input col
            float v = 0.0f;
            // causal input mask ig>=jg; ig>=jg && ig<H implies jg<W
            if (ig >= 0 && jg >= 0 && ig >= jg)
                v = xp[ig * Wc + jg];
            xs[i * WSTR + u] = v;
        }
        __syncthreads();

        // --- 168 x V_WMMA_F32_16X16X4_F32 ---
#pragma unroll
        for (int r = 0; r < Rr; ++r) {
#pragma unroll
            for (int t = 0; t < 7; ++t) {
                // B fragment: B(k,j) = w[r, u0+v - j], k = v + 2*half, j = col
                const int u0 = 4 * t + 2 * half;
                const int s0 = u0 - col;
                const int s1 = u0 + 1 - col;
                int s0c = s0 < 0 ? 0 : (s0 > 10 ? 10 : s0);
                int s1c = s1 < 0 ? 0 : (s1 > 10 ? 10 : s1);
                float b0 = lw[r * Ss + s0c];
                float b1 = lw[r * Ss + s1c];
                if (s0 != s0c) b0 = 0.0f;
                if (s1 != s1c) b1 = 0.0f;
                v2f bf = {b0, b1};

#pragma unroll
                for (int qt = 0; qt < 4; ++qt) {
                    // A fragment: A(m,k) = xs[m+r][16*qt + u0 + v], m = col
                    const v2f af = *reinterpret_cast<const v2f*>(
                        &xs[(col + r) * WSTR + 16 * qt + u0]);
                    acc[qt] = __builtin_amdgcn_wmma_f32_16x16x4_f32(
                        /*neg_a=*/false, af, /*neg_b=*/false, bf,
                        /*c_mod=*/(short)0, acc[qt],
                        /*reuse_a=*/false, /*reuse_b=*/false);
                }
            }
        }
    }

    // --- causal-masked store (zeros for masked / skipped region) ---
    // C/D layout: vgpr v -> M = v + 8*half, N = col
#pragma unroll
    for (int qt = 0; qt < 4; ++qt) {
        const int jq = q0 + 16 * qt + col;
#pragma unroll
        for (int v = 0; v < 8; ++v) {
            const int p = p0 + 8 * half + v;
            float val = (p >= jq) ? acc[qt][v] : 0.0f;
            yp[(size_t)p * Wc + jq] = val;
        }
    }
}

extern "C" void kernel_launch(void* const* d_in, const int* in_sizes, int n_in,
                              void* d_out, int out_size, void* d_ws, size_t ws_size,
                              hipStream_t stream) {
    (void)in_sizes; (void)n_in; (void)d_ws; (void)ws_size; (void)out_size;
    const float* x = (const float*)d_in[0];   // [N,C,H,W] fp32
    const float* w = (const float*)d_in[1];   // [C,R,S]  fp32
    float* y = (float*)d_out;                 // [N,C,H,W] fp32

    dim3 grid(Wc / 64, Hc / 16, Nb * Cc);     // 32 x 128 x 32 waves
    dim3 block(32, 1, 1);                     // one wave32 per workgroup
    dwconv_causal_wmma<<<grid, block, 0, stream>>>(x, w, y);
}